// GCNconv_1949915152808
// MI455X (gfx1250) — compile-verified
//
#include <hip/hip_runtime.h>

typedef __attribute__((ext_vector_type(2))) float v2f;
typedef __attribute__((ext_vector_type(8))) float v8f;

#define IN_DIM  256
#define OUT_DIM 256

// ---------------------------------------------------------------------------
// Kernel 1: h = x @ W   (x: [N,256] row-major f32, W: [256,256] row-major f32)
// Block: 256 threads = 8 waves. Block tile: 64 rows x 256 cols.
// Wave w: rows = blk*64 + (w>>2)*32, cols = (w&3)*64  -> 32x64 per wave:
// 8 accumulators (2 M-subtiles x 4 N-subtiles) of V_WMMA_F32_16X16X4_F32.
// Per k-step: 2 A-loads (8B each) + 8 B-loads (4B each) feed 8 WMMAs
// -> 16384 FLOP / 1536 B per wave-step (AI ~10.7).
// ---------------------------------------------------------------------------
__global__ __launch_bounds__(256) void gcn_gemm(const float* __restrict__ x,
                                                const float* __restrict__ W,
                                                float* __restrict__ h,
                                                int n_nodes) {
  const int tid     = threadIdx.x;
  const int wave    = tid >> 5;
  const int lane    = tid & 31;
  const int lane_lo = lane & 15;
  const int lane_hi = lane >> 4;          // 0 or 1 (half-wave)

  const int m_base = blockIdx.x * 64 + (wave >> 2) * 32;
  const int n_base = (wave & 3) * 64;

  v8f acc00 = {}, acc01 = {}, acc02 = {}, acc03 = {};   // rows m_base..+15
  v8f acc10 = {}, acc11 = {}, acc12 = {}, acc13 = {};   // rows m_base+16..+31

  // A (16x4 f32): lanes 0-15: (M=lane, K=k+0,k+1); lanes 16-31: (M, K=k+2,k+3)
  // Clamp rows so the (single) ragged tail block stays in-bounds; stores are
  // predicated instead, keeping EXEC all-ones around the WMMAs.
  int r0 = m_base + lane_lo;        if (r0 > n_nodes - 1) r0 = n_nodes - 1;
  int r1 = m_base + 16 + lane_lo;   if (r1 > n_nodes - 1) r1 = n_nodes - 1;
  const float* xrow0 = x + (size_t)r0 * IN_DIM + lane_hi * 2;
  const float* xrow1 = x + (size_t)r1 * IN_DIM + lane_hi * 2;

  // B (4x16 f32): VGPR0: K=k+0 (lanes 0-15) / K=k+2 (lanes 16-31), N=lane_lo
  //               VGPR1: K=k+1 / K=k+3
  const float* wbase = W + (size_t)(lane_hi * 2) * OUT_DIM + n_base + lane_lo;

  for (int k = 0; k < IN_DIM; k += 4) {
    v2f a0 = *(const v2f*)(xrow0 + k);
    v2f a1 = *(const v2f*)(xrow1 + k);
    const float* wk = wbase + (size_t)k * OUT_DIM;
    v2f b0, b1, b2, b3;
    b0.x = wk[0];  b0.y = wk[OUT_DIM];
    b1.x = wk[16]; b1.y = wk[OUT_DIM + 16];
    b2.x = wk[32]; b2.y = wk[OUT_DIM + 32];
    b3.x = wk[48]; b3.y = wk[OUT_DIM + 48];
    acc00 = __builtin_amdgcn_wmma_f32_16x16x4_f32(false, a0, false, b0, (short)0, acc00, false, false);
    acc01 = __builtin_amdgcn_wmma_f32_16x16x4_f32(false, a0, false, b1, (short)0, acc01, false, false);
    acc02 = __builtin_amdgcn_wmma_f32_16x16x4_f32(false, a0, false, b2, (short)0, acc02, false, false);
    acc03 = __builtin_amdgcn_wmma_f32_16x16x4_f32(false, a0, false, b3, (short)0, acc03, false, false);
    acc10 = __builtin_amdgcn_wmma_f32_16x16x4_f32(false, a1, false, b0, (short)0, acc10, false, false);
    acc11 = __builtin_amdgcn_wmma_f32_16x16x4_f32(false, a1, false, b1, (short)0, acc11, false, false);
    acc12 = __builtin_amdgcn_wmma_f32_16x16x4_f32(false, a1, false, b2, (short)0, acc12, false, false);
    acc13 = __builtin_amdgcn_wmma_f32_16x16x4_f32(false, a1, false, b3, (short)0, acc13, false, false);
  }

  // C/D (16x16 f32): VGPR i -> M = i + 8*lane_hi, N = lane_lo
  const int row0 = m_base + lane_hi * 8;          // M-subtile 0
  const int row1 = row0 + 16;                     // M-subtile 1
  float* hp0 = h + (size_t)row0 * OUT_DIM + n_base + lane_lo;
  float* hp1 = h + (size_t)row1 * OUT_DIM + n_base + lane_lo;
#pragma unroll
  for (int i = 0; i < 8; ++i) {
    if (row0 + i < n_nodes) {
      hp0[(size_t)i * OUT_DIM +  0] = acc00[i];
      hp0[(size_t)i * OUT_DIM + 16] = acc01[i];
      hp0[(size_t)i * OUT_DIM + 32] = acc02[i];
      hp0[(size_t)i * OUT_DIM + 48] = acc03[i];
    }
    if (row1 + i < n_nodes) {
      hp1[(size_t)i * OUT_DIM +  0] = acc10[i];
      hp1[(size_t)i * OUT_DIM + 16] = acc11[i];
      hp1[(size_t)i * OUT_DIM + 32] = acc12[i];
      hp1[(size_t)i * OUT_DIM + 48] = acc13[i];
    }
  }
}

// ---------------------------------------------------------------------------
// Kernel 2: out[n][c] = bias[c]   (bias folded in before atomic scatter)
// ---------------------------------------------------------------------------
__global__ __launch_bounds__(256) void gcn_init_out(const float* __restrict__ bias,
                                                    float* __restrict__ out,
                                                    int total) {
  int i = blockIdx.x * blockDim.x + threadIdx.x;
  if (i < total) out[i] = bias[i & (OUT_DIM - 1)];
}

// ---------------------------------------------------------------------------
// Kernel 3: out[src] += edge_val * h[dst]   — one wave per edge.
// Lane l covers channels l, l+32, ..., l+224: fully coalesced 128B loads of
// h[dst] (L2-resident: h = 102MB < 192MB L2). Scatter uses the hardware
// no-return f32 atomic via inline CDNA5 asm (global_atomic_add_f32, TH=0 ->
// non-returning, tracked on STOREcnt, RMW resolved in L2). All 8 gather
// loads + multiplies are done before the 8 atomics issue so the "memory"
// clobbers don't serialize load/atomic pairs.
// ---------------------------------------------------------------------------
__global__ __launch_bounds__(256) void gcn_scatter(const float* __restrict__ h,
                                                   const float* __restrict__ edge_val,
                                                   const int*   __restrict__ edge_src,
                                                   const int*   __restrict__ edge_dst,
                                                   float* __restrict__ out,
                                                   int n_edges) {
  const int lane   = threadIdx.x & 31;
  const int wid    = (blockIdx.x * blockDim.x + threadIdx.x) >> 5;
  const int nwaves = (gridDim.x * blockDim.x) >> 5;

  for (int e = wid; e < n_edges; e += nwaves) {
    const int   src = edge_src[e];
    const int   dst = edge_dst[e];
    const float val = edge_val[e];
    const float* hrow = h   + (size_t)dst * OUT_DIM;
    float*       orow = out + (size_t)src * OUT_DIM;

    float prod[8];
#pragma unroll
    for (int i = 0; i < 8; ++i)
      prod[i] = val * hrow[lane + i * 32];      // 8 coalesced 128B gathers

#pragma unroll
    for (int i = 0; i < 8; ++i) {
      float* p = orow + lane + i * 32;
      asm volatile("global_atomic_add_f32 %0, %1, off"
                   :
                   : "v"(p), "v"(prod[i])
                   : "memory");
    }
  }
}

// ---------------------------------------------------------------------------
extern "C" void kernel_launch(void* const* d_in, const int* in_sizes, int n_in,
                              void* d_out, int out_size, void* d_ws, size_t ws_size,
                              hipStream_t stream) {
  const float* x        = (const float*)d_in[0];
  const float* W        = (const float*)d_in[1];
  const float* bias     = (const float*)d_in[2];
  const float* edge_val = (const float*)d_in[3];
  const int*   edge_src = (const int*)d_in[4];
  const int*   edge_dst = (const int*)d_in[5];
  float* out = (float*)d_out;
  float* h   = (float*)d_ws;                      // 100000*256*4 = 102.4 MB

  const int n_nodes = in_sizes[0] / IN_DIM;       // 100000
  const int n_edges = in_sizes[3];                // 3200000
  const int total   = n_nodes * OUT_DIM;

  // 1) dense projection into workspace (WMMA f32), 64 rows per block
  hipLaunchKernelGGL(gcn_gemm, dim3((n_nodes + 63) / 64), dim3(256), 0, stream,
                     x, W, h, n_nodes);

  // 2) out = bias (broadcast)  — independent of (1), stream order is fine
  hipLaunchKernelGGL(gcn_init_out, dim3((total + 255) / 256), dim3(256), 0, stream,
                     bias, out, total);

  // 3) edge scatter with hw f32 atomics: one wave per edge
  hipLaunchKernelGGL(gcn_scatter, dim3((n_edges + 7) / 8), dim3(256), 0, stream,
                     h, edge_val, edge_src, edge_dst, out, n_edges);
}